// HybridLoss_85152021611306
// MI455X (gfx1250) — compile-verified
//
#include <hip/hip_runtime.h>
#include <math.h>

typedef float v2f __attribute__((ext_vector_type(2)));
typedef float v8f __attribute__((ext_vector_type(8)));

#define TPB 256
#define NBLK 1024
#define OVW 0.5f
#define EPS_ 1e-6f

// ---------------------------------------------------------------------------
// Reduce two quantities (pl, ov) across all 32 lanes of a wave using one
// V_WMMA_F32_16X16X4_F32.
//   A (16x4): lane L<16 -> A[L,0]=pl, A[L,1]=ov ; lane L>=16 -> A[L-16,2]=pl,
//             A[L-16,3]=ov  (per ISA 16x4 A layout, 2 VGPRs/lane)
//   B (4x16): B[0,n]=B[2,n]=(n<8), B[1,n]=B[3,n]=(n>=8)  (column selector)
//   D[m,n]  = (n<8) ? pl[m]+pl[m+16] : ov[m]+ov[m+16]
// Sum D's 8 VGPRs (rows m=0..7 in lanes 0-15, m=8..15 in lanes 16-31), then
// one shfl_xor(16) add finishes both sums: lanes n<8 hold total(pl),
// lanes 8<=n<16 hold total(ov). Must be called with the full wave active.
// ---------------------------------------------------------------------------
__device__ __forceinline__ void wave_reduce_pair(float pl, float ov,
                                                 float& tpl, float& tov) {
#if __has_builtin(__builtin_amdgcn_wmma_f32_16x16x4_f32)
  unsigned lane = threadIdx.x & 31u;
  v2f a; a[0] = pl; a[1] = ov;
  float selpl = ((lane & 15u) < 8u) ? 1.0f : 0.0f;
  v2f b; b[0] = selpl; b[1] = 1.0f - selpl;
  v8f c = {};
  v8f d = __builtin_amdgcn_wmma_f32_16x16x4_f32(
      /*neg_a=*/false, a, /*neg_b=*/false, b,
      /*c_mod=*/(short)0, c, /*reuse_a=*/false, /*reuse_b=*/false);
  float s = ((d[0] + d[1]) + (d[2] + d[3])) + ((d[4] + d[5]) + (d[6] + d[7]));
  s += __shfl_xor(s, 16, 32);
  tpl = __shfl(s, 0, 32);
  tov = __shfl(s, 8, 32);
#else
  for (int off = 16; off > 0; off >>= 1) {
    pl += __shfl_xor(pl, off, 32);
    ov += __shfl_xor(ov, off, 32);
  }
  tpl = pl; tov = ov;
#endif
}

__device__ __forceinline__ void quat_rot(float x, float y, float z, float w,
                                         float r[9]) {
  float xx = x * x, yy = y * y, zz = z * z;
  float xy = x * y, xz = x * z, yz = y * z;
  float xw = x * w, yw = y * w, zw = z * w;
  r[0] = 1.f - 2.f * (yy + zz); r[1] = 2.f * (xy + zw); r[2] = 2.f * (xz - yw);
  r[3] = 2.f * (xy - zw); r[4] = 1.f - 2.f * (xx + zz); r[5] = 2.f * (yz + xw);
  r[6] = 2.f * (xz + yw); r[7] = 2.f * (yz - xw); r[8] = 1.f - 2.f * (xx + yy);
}

// Per-row fused smooth-L1 sum + overlap.
__device__ __forceinline__ void row_loss(const float* __restrict__ P,
                                         const float* __restrict__ T,
                                         long long i, float& sl1, float& ovl) {
  // rows are 40 B -> five 8-B-aligned float2 loads per tensor (global_load_b64)
  const float2* pr = reinterpret_cast<const float2*>(P + i * 10);
  const float2* tr = reinterpret_cast<const float2*>(T + i * 10);
  float p[10], t[10];
#pragma unroll
  for (int j = 0; j < 5; ++j) {
    float2 v = pr[j]; p[2 * j] = v.x; p[2 * j + 1] = v.y;
    float2 w = tr[j]; t[2 * j] = w.x; t[2 * j + 1] = w.y;
  }

  // smooth L1 (sum over 10 params)
  float s = 0.f;
#pragma unroll
  for (int j = 0; j < 10; ++j) {
    float d = p[j] - t[j];
    float ad = fabsf(d);
    s += (ad < 1.f) ? 0.5f * d * d : (ad - 0.5f);
  }
  sl1 = s;

  // normalized quats -> rotation matrices
  float pn = 1.0f / sqrtf(p[6] * p[6] + p[7] * p[7] + p[8] * p[8] + p[9] * p[9]);
  float gn = 1.0f / sqrtf(t[6] * t[6] + t[7] * t[7] + t[8] * t[8] + t[9] * t[9]);
  float pR[9], gR[9];
  quat_rot(p[6] * pn, p[7] * pn, p[8] * pn, p[9] * pn, pR);
  quat_rot(t[6] * gn, t[7] * gn, t[8] * gn, t[9] * gn, gR);

  float dRR = 0.f;
#pragma unroll
  for (int k = 0; k < 9; ++k) dRR += pR[k] * gR[k];
  float rot_align = fminf(fmaxf(dRR * (1.f / 3.f), 0.f), 1.f);

  float prod = 1.f;
#pragma unroll
  for (int ax = 0; ax < 3; ++ax) {
    float pe = fabsf(pR[3 * ax + 0] * (p[3] * 0.5f) +
                     pR[3 * ax + 1] * (p[4] * 0.5f) +
                     pR[3 * ax + 2] * (p[5] * 0.5f));
    float ge = fabsf(gR[3 * ax + 0] * (t[3] * 0.5f) +
                     gR[3 * ax + 1] * (t[4] * 0.5f) +
                     gR[3 * ax + 2] * (t[5] * 0.5f));
    float cd = fabsf(p[ax] - t[ax]);
    float oa = fmaxf(fminf(pe, ge) * 2.f - cd, 0.f) / (pe + ge + EPS_);
    prod *= oa;
  }
  ovl = fminf(fmaxf(prod * rot_align, 0.f), 1.f);
}

__global__ void loss_partial(const float* __restrict__ preds,
                             const float* __restrict__ targets,
                             float* __restrict__ part, int nrows) {
  float pl = 0.f, ov = 0.f;
  for (long long i = blockIdx.x * (long long)blockDim.x + threadIdx.x;
       i < nrows; i += (long long)gridDim.x * blockDim.x) {
    float s, o;
    row_loss(preds, targets, i, s, o);
    pl += s; ov += o;
  }
  __shared__ float spl[TPB], sov[TPB];
  spl[threadIdx.x] = pl;
  sov[threadIdx.x] = ov;
  __syncthreads();
  if (threadIdx.x < 32) {  // wave-uniform: wave 0 keeps EXEC all-ones
    float a = 0.f, b = 0.f;
#pragma unroll
    for (int k = threadIdx.x; k < TPB; k += 32) { a += spl[k]; b += sov[k]; }
    float tpl, tov;
    wave_reduce_pair(a, b, tpl, tov);
    if (threadIdx.x == 0) {
      part[2 * blockIdx.x + 0] = tpl;
      part[2 * blockIdx.x + 1] = tov;
    }
  }
}

__global__ void loss_finalize(const float* __restrict__ part,
                              float* __restrict__ out, int nparts, int nrows) {
  float pl = 0.f, ov = 0.f;
  for (int k = threadIdx.x; k < nparts; k += blockDim.x) {
    pl += part[2 * k + 0];
    ov += part[2 * k + 1];
  }
  __shared__ float spl[TPB], sov[TPB];
  spl[threadIdx.x] = pl;
  sov[threadIdx.x] = ov;
  __syncthreads();
  if (threadIdx.x < 32) {
    float a = 0.f, b = 0.f;
#pragma unroll
    for (int k = threadIdx.x; k < TPB; k += 32) { a += spl[k]; b += sov[k]; }
    float tpl, tov;
    wave_reduce_pair(a, b, tpl, tov);
    if (threadIdx.x == 0) {
      float param_loss = tpl / ((float)nrows * 10.0f);
      float mean_ov = tov / (float)nrows;
      out[0] = param_loss + OVW * (1.0f - mean_ov);
    }
  }
}

extern "C" void kernel_launch(void* const* d_in, const int* in_sizes, int n_in,
                              void* d_out, int out_size, void* d_ws,
                              size_t ws_size, hipStream_t stream) {
  const float* preds = (const float*)d_in[0];
  const float* targets = (const float*)d_in[1];
  int nrows = in_sizes[0] / 10;
  float* part = (float*)d_ws;  // 2*NBLK floats, fully overwritten each call

  loss_partial<<<NBLK, TPB, 0, stream>>>(preds, targets, part, nrows);
  loss_finalize<<<1, TPB, 0, stream>>>(part, (float*)d_out, NBLK, nrows);
}